// AttributeHead_2671469658608
// MI455X (gfx1250) — compile-verified
//
#include <hip/hip_runtime.h>

typedef __attribute__((ext_vector_type(2))) float v2f;
typedef __attribute__((ext_vector_type(8))) float v8f;

#define NROI 1024
#define OC   256
#define SSZ  14
#define BINS 49

// ---------------------------------------------------------------------------
// Kernel 1: fused ROI-align (on raw features) + 1x1 conv via WMMA f32 16x16x4.
// One block (256 threads / 8 waves) per roi. Conv-weight K-chunk is fetched
// with GLOBAL_LOAD_ASYNC_TO_LDS_B128 so its latency hides behind the bilinear
// gather loop. Bias is added scaled by the per-bin valid-sample fraction.
// ---------------------------------------------------------------------------
__global__ __launch_bounds__(256) void roi_conv_kernel(
    const float* __restrict__ feat0, const float* __restrict__ feat1,
    const float* __restrict__ feat2, const float* __restrict__ rois,
    const float* __restrict__ cw0, const float* __restrict__ cb0,
    const float* __restrict__ cw1, const float* __restrict__ cb1,
    const float* __restrict__ cw2, const float* __restrict__ cb2,
    float* __restrict__ flat)
{
  __shared__ float sA[OC][36];       // conv weight K-chunk; stride 36 dwords
                                     // (144B: 16B-aligned rows, conflict-free)
  __shared__ float sPool[32][64];    // pooled raw features, N padded to 64
  __shared__ int   sYlo[SSZ], sYhi[SSZ], sVy[SSZ];
  __shared__ float sFy[SSZ];
  __shared__ int   sXlo[SSZ], sXhi[SSZ], sVx[SSZ];
  __shared__ float sFx[SSZ];
  __shared__ float sFrac[BINS];
  __shared__ float sCb[OC];
  __shared__ float sGeo[4];          // x1, y1, rw, rh
  __shared__ int   sMeta[5];         // lvl, batch, C, H, W

  const int tid = threadIdx.x;
  const int n = blockIdx.x;

  if (tid == 0) {
    float bi  = rois[n*5 + 0];
    float rx1 = rois[n*5 + 1], ry1 = rois[n*5 + 2];
    float rx2 = rois[n*5 + 3], ry2 = rois[n*5 + 4];
    float bw = rx2 - rx1, bh = ry2 - ry1;
    float kf = floorf(4.0f + log2f(sqrtf(bw*bh) / 224.0f));
    int lvl = (int)fminf(fmaxf(kf, 3.0f), 5.0f) - 3;
    float scale = 1.0f / (float)(8 << lvl);
    float x1 = rx1*scale - 0.5f;
    float y1 = ry1*scale - 0.5f;
    sGeo[0] = x1; sGeo[1] = y1;
    sGeo[2] = rx2*scale - 0.5f - x1;
    sGeo[3] = ry2*scale - 0.5f - y1;
    sMeta[0] = lvl;
    sMeta[1] = (int)bi;
    sMeta[2] = (lvl==0) ? 128 : ((lvl==1) ? 256 : 512);
    sMeta[3] = 80 >> lvl;
    sMeta[4] = 80 >> lvl;
  }
  __syncthreads();

  const int lvl = sMeta[0];
  const float* __restrict__ feat = (lvl==0) ? feat0 : ((lvl==1) ? feat1 : feat2);
  const float* __restrict__ cw   = (lvl==0) ? cw0   : ((lvl==1) ? cw1   : cw2);
  const float* __restrict__ cb   = (lvl==0) ? cb0   : ((lvl==1) ? cb1   : cb2);
  const int C = sMeta[2], H = sMeta[3], W = sMeta[4];
  const int bimg = sMeta[1];

  // per-axis bilinear setup (exactly mirrors reference axis())
  if (tid < 2*SSZ) {
    bool isY = tid < SSZ;
    int  i   = isY ? tid : tid - SSZ;
    int  size = isY ? H : W;
    float p = ((float)i + 0.5f) / (float)SSZ;
    float v = isY ? (sGeo[1] + sGeo[3]*p) : (sGeo[0] + sGeo[2]*p);
    int valid = (v >= -1.0f) && (v <= (float)size);
    v = fmaxf(v, 0.0f);
    float lo = fminf(floorf(v), (float)(size-1));
    float hi = fminf(lo + 1.0f, (float)(size-1));
    float f = v - lo;
    if (isY) { sYlo[i]=(int)lo; sYhi[i]=(int)hi; sFy[i]=f; sVy[i]=valid; }
    else     { sXlo[i]=(int)lo; sXhi[i]=(int)hi; sFx[i]=f; sVx[i]=valid; }
  }
  sCb[tid] = cb[tid];
  // zero N-pad columns once (never written afterwards)
  for (int i = tid; i < 32*15; i += 256) sPool[i/15][BINS + (i%15)] = 0.0f;
  __syncthreads();
  if (tid < BINS) {
    int by = tid/7, bx = tid%7, cnt = 0;
    #pragma unroll
    for (int sy=0; sy<2; sy++)
      #pragma unroll
      for (int sx=0; sx<2; sx++)
        cnt += (sVy[by*2+sy] & sVx[bx*2+sx]);
    sFrac[tid] = (float)cnt * 0.25f;
  }

  const int wave = tid >> 5;
  const int lane = tid & 31;
  const int laneHi = lane >> 4;
  const int laneLo = lane & 15;

  // async-copy lane mapping: thread -> (row = tid>>3 [+32j], 4-float segment)
  const int rr  = tid >> 3;          // 0..31
  const int seg = (tid & 7) << 2;    // 0,4,...,28
  const unsigned ldsA = (unsigned)(size_t)&sA[rr][seg];

  v8f acc[2][4] = {};

  const size_t plane = (size_t)H * (size_t)W;
  const float* __restrict__ fbase = feat + (size_t)bimg * (size_t)C * plane;

  for (int kb = 0; kb < C; kb += 32) {
    __syncthreads();
    // async-stage conv-weight chunk cw[0:256, kb:kb+32] into LDS (8 x B128,
    // each covering 32 rows x 32 cols across the block's lanes)
    #pragma unroll
    for (int j = 0; j < 8; j++) {
      unsigned goff = (unsigned)((((rr + j*32) * C) + kb + seg) * 4);
      asm volatile("global_load_async_to_lds_b128 %0, %1, %2"
                   :: "v"(ldsA + (unsigned)(j * 32 * 144)), "v"(goff), "s"(cw)
                   : "memory");
    }
    // gather + pool 32 raw channels into sPool[ch][bin] (hides async latency)
    for (int t = tid; t < 32*BINS; t += 256) {
      int cl = t / BINS, bin = t - cl*BINS;
      int by = bin / 7, bx = bin - by*7;
      const float* __restrict__ pl = fbase + (size_t)(kb + cl) * plane;
      float a = 0.0f;
      #pragma unroll
      for (int sy = 0; sy < 2; sy++) {
        int iy = by*2 + sy;
        if (!sVy[iy]) continue;
        float fy = sFy[iy];
        const float* rlo = pl + (size_t)sYlo[iy] * W;
        const float* rhi = pl + (size_t)sYhi[iy] * W;
        #pragma unroll
        for (int sx = 0; sx < 2; sx++) {
          int ix = bx*2 + sx;
          if (!sVx[ix]) continue;
          float fx = sFx[ix];
          int xl = sXlo[ix], xh = sXhi[ix];
          a += (1.0f-fy)*((1.0f-fx)*rlo[xl] + fx*rlo[xh])
             +        fy*((1.0f-fx)*rhi[xl] + fx*rhi[xh]);
        }
      }
      sPool[cl][bin] = a * 0.25f;
    }
    asm volatile("s_wait_asynccnt 0" ::: "memory");
    __syncthreads();
    // WMMA: out[256x64] += cw_chunk[256x32] x pooled[32x64]
    #pragma unroll
    for (int kk = 0; kk < 32; kk += 4) {
      int krow = kk + 2*laneHi;
      int o0 = (wave*2 + 0)*16 + laneLo;
      int o1 = (wave*2 + 1)*16 + laneLo;
      v2f a0, a1;
      a0.x = sA[o0][krow]; a0.y = sA[o0][krow+1];
      a1.x = sA[o1][krow]; a1.y = sA[o1][krow+1];
      #pragma unroll
      for (int nt = 0; nt < 4; nt++) {
        v2f b;
        b.x = sPool[krow][nt*16 + laneLo];
        b.y = sPool[krow+1][nt*16 + laneLo];
        acc[0][nt] = __builtin_amdgcn_wmma_f32_16x16x4_f32(
            false, a0, false, b, (short)0, acc[0][nt], false, false);
        acc[1][nt] = __builtin_amdgcn_wmma_f32_16x16x4_f32(
            false, a1, false, b, (short)0, acc[1][nt], false, false);
      }
    }
  }

  // epilogue: bias * valid-fraction, write flat[n, o*49 + bin]
  float* __restrict__ outp = flat + (size_t)n * (OC*BINS);
  #pragma unroll
  for (int mi = 0; mi < 2; mi++) {
    int o16 = (wave*2 + mi)*16;
    #pragma unroll
    for (int nt = 0; nt < 4; nt++) {
      int bin = nt*16 + laneLo;
      if (bin < BINS) {
        float fr = sFrac[bin];
        #pragma unroll
        for (int r = 0; r < 8; r++) {
          int o = o16 + r + 8*laneHi;
          outp[o*BINS + bin] = acc[mi][nt][r] + sCb[o]*fr;
        }
      }
    }
  }
}

// ---------------------------------------------------------------------------
// Kernel 2: tiled GEMM Y = A[M,K] * W[N,K]^T (+bias, optional BN+ReLU).
// 64x64 block tile, K chunk 32, 8 waves x 2 acc tiles, WMMA f32 16x16x4.
// Double-buffered GLOBAL_LOAD_ASYNC_TO_LDS_B128 staging: next K-chunk's
// tiles stream into LDS while the current chunk's WMMAs execute.
// ---------------------------------------------------------------------------
__global__ __launch_bounds__(256) void fc_kernel(
    const float* __restrict__ A, const float* __restrict__ Wt,
    const float* __restrict__ bias,
    const float* __restrict__ g, const float* __restrict__ be,
    const float* __restrict__ rm, const float* __restrict__ rv,
    float* __restrict__ Y, int M, int N, int K, int bnrelu)
{
  __shared__ float sA[2][64][36];   // row-major [m][k], stride 36 (144B rows)
  __shared__ float sW[2][64][36];   // row-major [n][k]
  const int tid = threadIdx.x;
  const int wave = tid >> 5, lane = tid & 31;
  const int laneHi = lane >> 4, laneLo = lane & 15;
  const int mBase = blockIdx.x * 64;
  const int nBase = blockIdx.y * 64;

  const int mt = wave & 3;
  const int ntBase = (wave >> 2) * 2;

  // async-copy lane mapping
  const int rr  = tid >> 3;          // 0..31
  const int seg = (tid & 7) << 2;    // 0,4,...,28
  const unsigned ldsA = (unsigned)(size_t)&sA[0][rr][seg];
  const unsigned ldsW = (unsigned)(size_t)&sW[0][rr][seg];
  const unsigned bufStride = 64u * 144u;          // 9216 B per buffer
  const unsigned rowBlk    = 32u * 144u;          // +32 rows

  const unsigned offA0 = (unsigned)(((size_t)(mBase + rr     )*K + seg) * 4);
  const unsigned offA1 = (unsigned)(((size_t)(mBase + rr + 32)*K + seg) * 4);
  const unsigned offW0 = (unsigned)(((size_t)(nBase + rr     )*K + seg) * 4);
  const unsigned offW1 = (unsigned)(((size_t)(nBase + rr + 32)*K + seg) * 4);

#define FC_PREFETCH(buf, kbyte)                                                 \
  do {                                                                          \
    unsigned bo = (unsigned)(buf) * bufStride;                                  \
    asm volatile("global_load_async_to_lds_b128 %0, %1, %2"                     \
                 :: "v"(ldsA + bo),          "v"(offA0 + (kbyte)), "s"(A)       \
                 : "memory");                                                   \
    asm volatile("global_load_async_to_lds_b128 %0, %1, %2"                     \
                 :: "v"(ldsA + bo + rowBlk), "v"(offA1 + (kbyte)), "s"(A)       \
                 : "memory");                                                   \
    asm volatile("global_load_async_to_lds_b128 %0, %1, %2"                     \
                 :: "v"(ldsW + bo),          "v"(offW0 + (kbyte)), "s"(Wt)      \
                 : "memory");                                                   \
    asm volatile("global_load_async_to_lds_b128 %0, %1, %2"                     \
                 :: "v"(ldsW + bo + rowBlk), "v"(offW1 + (kbyte)), "s"(Wt)      \
                 : "memory");                                                   \
  } while (0)

  v8f acc[2] = {};

  const int nChunks = K >> 5;
  FC_PREFETCH(0, 0u);
  for (int ci = 0; ci < nChunks; ci++) {
    const int buf = ci & 1;
    if (ci + 1 < nChunks) {
      FC_PREFETCH(1 - buf, (unsigned)(ci + 1) * 128u);
      asm volatile("s_wait_asynccnt 4" ::: "memory");  // oldest 4 (this buf) done
    } else {
      asm volatile("s_wait_asynccnt 0" ::: "memory");
    }
    __syncthreads();
    #pragma unroll
    for (int kk = 0; kk < 32; kk += 4) {
      int krow = kk + 2*laneHi;
      v2f a;
      a.x = sA[buf][mt*16 + laneLo][krow];
      a.y = sA[buf][mt*16 + laneLo][krow+1];
      #pragma unroll
      for (int i = 0; i < 2; i++) {
        int nt = ntBase + i;
        v2f b;
        b.x = sW[buf][nt*16 + laneLo][krow];
        b.y = sW[buf][nt*16 + laneLo][krow+1];
        acc[i] = __builtin_amdgcn_wmma_f32_16x16x4_f32(
            false, a, false, b, (short)0, acc[i], false, false);
      }
    }
    __syncthreads();
  }
#undef FC_PREFETCH

  #pragma unroll
  for (int i = 0; i < 2; i++) {
    int nIdx = nBase + (ntBase+i)*16 + laneLo;
    float bia = bias[nIdx];
    float scale = 1.0f, shift = 0.0f;
    if (bnrelu) {
      scale = rsqrtf(rv[nIdx] + 1e-5f) * g[nIdx];
      shift = be[nIdx] - rm[nIdx]*scale;
    }
    #pragma unroll
    for (int r = 0; r < 8; r++) {
      int m = mBase + mt*16 + r + 8*laneHi;
      float v = acc[i][r] + bia;
      if (bnrelu) v = fmaxf(v*scale + shift, 0.0f);
      Y[(size_t)m*N + nIdx] = v;
    }
  }
}

// ---------------------------------------------------------------------------
// Kernel 3: final tiny FC (N=26) — VALU dot products.
// ---------------------------------------------------------------------------
__global__ __launch_bounds__(256) void fc3_kernel(
    const float* __restrict__ h2, const float* __restrict__ fw3,
    const float* __restrict__ fb3, float* __restrict__ out)
{
  int idx = blockIdx.x * blockDim.x + threadIdx.x;
  if (idx >= NROI * 26) return;
  int n = idx / 26, j = idx - n*26;
  const float* a = h2  + (size_t)n * 256;
  const float* w = fw3 + (size_t)j * 256;
  float s = fb3[j];
  #pragma unroll 8
  for (int k = 0; k < 256; k++) s += a[k] * w[k];
  out[idx] = s;
}

extern "C" void kernel_launch(void* const* d_in, const int* in_sizes, int n_in,
                              void* d_out, int out_size, void* d_ws, size_t ws_size,
                              hipStream_t stream) {
  const float* feat0 = (const float*)d_in[0];
  const float* feat1 = (const float*)d_in[1];
  const float* feat2 = (const float*)d_in[2];
  const float* rois  = (const float*)d_in[3];
  const float* cw0 = (const float*)d_in[4];
  const float* cb0 = (const float*)d_in[5];
  const float* cw1 = (const float*)d_in[6];
  const float* cb1 = (const float*)d_in[7];
  const float* cw2 = (const float*)d_in[8];
  const float* cb2 = (const float*)d_in[9];
  const float* fw1 = (const float*)d_in[10];
  const float* fb1 = (const float*)d_in[11];
  const float* g1  = (const float*)d_in[12];
  const float* be1 = (const float*)d_in[13];
  const float* rm1 = (const float*)d_in[14];
  const float* rv1 = (const float*)d_in[15];
  const float* fw2 = (const float*)d_in[16];
  const float* fb2 = (const float*)d_in[17];
  const float* g2  = (const float*)d_in[18];
  const float* be2 = (const float*)d_in[19];
  const float* rm2 = (const float*)d_in[20];
  const float* rv2 = (const float*)d_in[21];
  const float* fw3 = (const float*)d_in[22];
  const float* fb3 = (const float*)d_in[23];

  float* flat = (float*)d_ws;                        // 1024*12544
  float* h1   = flat + (size_t)NROI * 12544;         // 1024*512
  float* h2   = h1   + (size_t)NROI * 512;           // 1024*256

  roi_conv_kernel<<<dim3(NROI), dim3(256), 0, stream>>>(
      feat0, feat1, feat2, rois, cw0, cb0, cw1, cb1, cw2, cb2, flat);
  fc_kernel<<<dim3(16, 8), dim3(256), 0, stream>>>(
      flat, fw1, fb1, g1, be1, rm1, rv1, h1, NROI, 512, 12544, 1);
  fc_kernel<<<dim3(16, 4), dim3(256), 0, stream>>>(
      h1, fw2, fb2, g2, be2, rm2, rv2, h2, NROI, 256, 512, 1);
  fc3_kernel<<<dim3((NROI*26 + 255)/256), dim3(256), 0, stream>>>(
      h2, fw3, fb3, (float*)d_out);
}